// EdgeAttentionLayer_50938312131110
// MI455X (gfx1250) — compile-verified
//
#include <hip/hip_runtime.h>
#include <hip/hip_bf16.h>
#include <stdint.h>

#define N_NODES 50000
#define N_EDGES 400000
#define N_E2E   1600000
#define HD      128
#define NEG_SLOPE 0.2f

typedef __attribute__((ext_vector_type(2))) float v2f;
typedef __attribute__((ext_vector_type(8))) float v8f;

__device__ __forceinline__ float lrelu(float x) { return x >= 0.f ? x : NEG_SLOPE * x; }

// order-preserving map float -> uint for atomicMax
__device__ __forceinline__ unsigned fmap(float f) {
    unsigned u = __float_as_uint(f);
    return (u & 0x80000000u) ? ~u : (u | 0x80000000u);
}
__device__ __forceinline__ float funmap(unsigned u) {
    unsigned b = (u & 0x80000000u) ? (u ^ 0x80000000u) : ~u;
    return __uint_as_float(b);
}

// ---------------- zero init ----------------
__global__ void zero_f4_k(float* __restrict__ p, long n4) {
    long i = (long)blockIdx.x * blockDim.x + threadIdx.x;
    if (i >= n4) return;
    *(float4*)(p + 4 * i) = make_float4(0.f, 0.f, 0.f, 0.f);
}

// ---------------- WMMA fp32 GEMM: C[M,N] = A[M,K] * W[N,K]^T ----------------
// one wave per 16x16 output tile, K-loop in steps of 4 via v_wmma_f32_16x16x4_f32
__global__ void wmma_gemm_bt_k(const float* __restrict__ A,
                               const float* __restrict__ W,
                               float* __restrict__ C,
                               int Mtiles, int Ntiles, int K, int N) {
    int lane = threadIdx.x & 31;
    int wave = threadIdx.x >> 5;
    int tile = blockIdx.x * (blockDim.x >> 5) + wave;
    if (tile >= Mtiles * Ntiles) return;   // uniform per wave -> EXEC stays all-1 for WMMA
    int mt = tile / Ntiles, nt = tile % Ntiles;
    int r16 = lane & 15;
    int kof = (lane >> 4) << 1;            // lanes 0-15: K pair {0,1}; lanes 16-31: {2,3}
    const float* Ap = A + (size_t)(mt * 16 + r16) * K + kof;
    const float* Wp = W + (size_t)(nt * 16 + r16) * K + kof;
    v8f acc = {0.f, 0.f, 0.f, 0.f, 0.f, 0.f, 0.f, 0.f};
#pragma unroll 8
    for (int k = 0; k < K; k += 4) {
        v2f a = *(const v2f*)(Ap + k);
        v2f b = *(const v2f*)(Wp + k);
        acc = __builtin_amdgcn_wmma_f32_16x16x4_f32(false, a, false, b,
                                                    (short)0, acc, false, false);
    }
    // D layout: VGPR i -> row mt*16 + i + (lane/16)*8, col nt*16 + lane%16
    int row0 = mt * 16 + ((lane >> 4) << 3);
    int col  = nt * 16 + r16;
#pragma unroll
    for (int i = 0; i < 8; ++i)
        C[(size_t)(row0 + i) * N + col] = acc[i];
}

// ---------------- per-row dot products ----------------
__global__ void rowdot1_k(const float* __restrict__ M, const float* __restrict__ vsh,
                          float* __restrict__ outv, int rows) {
    int gtid = blockIdx.x * blockDim.x + threadIdx.x;
    int row = gtid >> 5, lane = threadIdx.x & 31;
    if (row >= rows) return;
    float4 m4 = *(const float4*)(M + (size_t)row * HD + lane * 4);
    float4 a4 = *(const float4*)(vsh + lane * 4);
    float s = m4.x * a4.x + m4.y * a4.y + m4.z * a4.z + m4.w * a4.w;
#pragma unroll
    for (int o = 16; o; o >>= 1) s += __shfl_xor(s, o, 32);
    if (lane == 0) outv[row] = s;
}

__global__ void rowdot3_k(const float* __restrict__ G, const float* __restrict__ a1,
                          const float* __restrict__ b1, const float* __restrict__ b2,
                          float* __restrict__ p, float* __restrict__ r,
                          float* __restrict__ t, int rows) {
    int gtid = blockIdx.x * blockDim.x + threadIdx.x;
    int row = gtid >> 5, lane = threadIdx.x & 31;
    if (row >= rows) return;
    float4 g4 = *(const float4*)(G + (size_t)row * HD + lane * 4);
    float4 A = *(const float4*)(a1 + lane * 4);
    float4 B = *(const float4*)(b1 + lane * 4);
    float4 Cc = *(const float4*)(b2 + lane * 4);
    float s1 = g4.x * A.x + g4.y * A.y + g4.z * A.z + g4.w * A.w;
    float s2 = g4.x * B.x + g4.y * B.y + g4.z * B.z + g4.w * B.w;
    float s3 = g4.x * Cc.x + g4.y * Cc.y + g4.z * Cc.z + g4.w * Cc.w;
#pragma unroll
    for (int o = 16; o; o >>= 1) {
        s1 += __shfl_xor(s1, o, 32);
        s2 += __shfl_xor(s2, o, 32);
        s3 += __shfl_xor(s3, o, 32);
    }
    if (lane == 0) { p[row] = s1; r[row] = s2; t[row] = s3; }
}

// ---------------- node->edge attention (segments of exactly 2) ----------------
__global__ void node_edge_k(const float* __restrict__ h, const float* __restrict__ p,
                            const float* __restrict__ q, const int* __restrict__ eidx,
                            float* __restrict__ out) {
    int gtid = blockIdx.x * blockDim.x + threadIdx.x;
    int e = gtid >> 5, lane = threadIdx.x & 31;
    if (e >= N_EDGES) return;
    int u = eidx[e], v = eidx[N_EDGES + e];
    float pe = p[e];
    float su = lrelu(pe + q[u]);
    float sv = lrelu(pe + q[v]);
    float m = fmaxf(su, sv);
    float eu = __expf(su - m), ev = __expf(sv - m);
    float inv = 0.5f / (eu + ev);           // softmax / cnt(=2)
    float4 hu = *(const float4*)(h + (size_t)u * HD + lane * 4);
    float4 hv = *(const float4*)(h + (size_t)v * HD + lane * 4);
    float4 o;
    o.x = fmaxf(0.f, (eu * hu.x + ev * hv.x) * inv);
    o.y = fmaxf(0.f, (eu * hu.y + ev * hv.y) * inv);
    o.z = fmaxf(0.f, (eu * hu.z + ev * hv.z) * inv);
    o.w = fmaxf(0.f, (eu * hu.w + ev * hv.w) * inv);
    *(float4*)(out + (size_t)e * 256 + lane * 4) = o;
}

// ---------------- edge->edge: pass A, segment max + count ----------------
__global__ void e2e_max_cnt_k(const float* __restrict__ r, const float* __restrict__ t,
                              const int* __restrict__ eei, unsigned* __restrict__ mbuf,
                              float* __restrict__ cnt) {
    int k = blockIdx.x * blockDim.x + threadIdx.x;
    if (k >= N_E2E) return;
    int s = eei[k], nb = eei[N_E2E + k];
    float sc = lrelu(r[s] + t[nb]);
    atomicMax(mbuf + s, fmap(sc));
    atomicAdd(cnt + s, 1.0f);
}

// ---------------- edge->edge: pass B, exp/denom + weighted scatter into out ----------------
__global__ void e2e_accum_k(const float* __restrict__ G, const float* __restrict__ r,
                            const float* __restrict__ t, const int* __restrict__ eei,
                            const unsigned* __restrict__ mbuf, float* __restrict__ denom,
                            float* __restrict__ out) {
    int gtid = blockIdx.x * blockDim.x + threadIdx.x;
    int k = gtid >> 5, lane = threadIdx.x & 31;
    if (k >= N_E2E) return;
    int s = eei[k], nb = eei[N_E2E + k];
    float sc = lrelu(r[s] + t[nb]);
    float ex = __expf(sc - funmap(mbuf[s]));
    if (lane == 0) atomicAdd(denom + s, ex);
    float4 g4 = *(const float4*)(G + (size_t)nb * HD + lane * 4);
    float* o = out + (size_t)s * 256 + 128 + lane * 4;
    atomicAdd(o + 0, ex * g4.x);
    atomicAdd(o + 1, ex * g4.y);
    atomicAdd(o + 2, ex * g4.z);
    atomicAdd(o + 3, ex * g4.w);
}

// ---------------- edge->edge: pass C, normalize in place ----------------
__global__ void e2e_final_k(const float* __restrict__ denom, const float* __restrict__ cnt,
                            float* __restrict__ out) {
    int gtid = blockIdx.x * blockDim.x + threadIdx.x;
    int e = gtid >> 5, lane = threadIdx.x & 31;
    if (e >= N_EDGES) return;
    float d = denom[e];
    float scale = (d > 0.f) ? 1.0f / (d * fmaxf(cnt[e], 1.0f)) : 0.0f;
    float* o = out + (size_t)e * 256 + 128 + lane * 4;
    float4 v = *(float4*)o;
    v.x = fmaxf(0.f, v.x * scale);
    v.y = fmaxf(0.f, v.y * scale);
    v.z = fmaxf(0.f, v.z * scale);
    v.w = fmaxf(0.f, v.w * scale);
    *(float4*)o = v;
}

extern "C" void kernel_launch(void* const* d_in, const int* in_sizes, int n_in,
                              void* d_out, int out_size, void* d_ws, size_t ws_size,
                              hipStream_t stream) {
    const float* x_node    = (const float*)d_in[0];
    const float* edge_attr = (const float*)d_in[1];
    const int*   eidx      = (const int*)d_in[2];   // (2, N_EDGES) row-major
    const int*   eei       = (const int*)d_in[3];   // (2, N_E2E) row-major
    const float* Wn        = (const float*)d_in[4]; // (128,128)
    const float* We        = (const float*)d_in[5]; // (128,64)
    const float* an        = (const float*)d_in[6]; // (256,)
    const float* ae        = (const float*)d_in[7]; // (256,)
    float* out = (float*)d_out;

    float* ws = (float*)d_ws;
    size_t off = 0;
    float* h = ws + off;  off += (size_t)N_NODES * HD;   // 6.4M
    float* g = ws + off;  off += (size_t)N_EDGES * HD;   // 51.2M
    float* q = ws + off;  off += 50048;                  // padded N_NODES
    float* p = ws + off;  off += N_EDGES;
    float* r = ws + off;  off += N_EDGES;
    float* t = ws + off;  off += N_EDGES;
    float* aux = ws + off;                               // mbuf|denom|cnt contiguous
    unsigned* mbuf = (unsigned*)aux;
    float* denom = aux + N_EDGES;
    float* cnt   = aux + 2 * (size_t)N_EDGES;

    const int B = 256;

    // zero d_out (also serves as edge-edge accumulation buffer) and atomic buffers
    long out_n4 = (long)N_EDGES * 256 / 4;
    zero_f4_k<<<(int)((out_n4 + B - 1) / B), B, 0, stream>>>(out, out_n4);
    long aux_n4 = 3L * N_EDGES / 4;
    zero_f4_k<<<(int)((aux_n4 + B - 1) / B), B, 0, stream>>>(aux, aux_n4);

    // GEMMs via fp32 WMMA: 8 waves/block, one 16x16 tile per wave
    wmma_gemm_bt_k<<<3125, B, 0, stream>>>(x_node, Wn, h, 3125, 8, 128, 128);
    wmma_gemm_bt_k<<<25000, B, 0, stream>>>(edge_attr, We, g, 25000, 8, 64, 128);

    // scalar projections: q = h . an[128:], p = g . an[:128], r = g . ae[:128], t = g . ae[128:]
    rowdot1_k<<<(N_NODES * 32 + B - 1) / B, B, 0, stream>>>(h, an + HD, q, N_NODES);
    rowdot3_k<<<(N_EDGES * 32 + B - 1) / B, B, 0, stream>>>(g, an, ae, ae + HD, p, r, t, N_EDGES);

    // node -> edge half (out[:, :128])
    node_edge_k<<<(N_EDGES * 32 + B - 1) / B, B, 0, stream>>>(h, p, q, eidx, out);

    // edge -> edge half (out[:, 128:])
    e2e_max_cnt_k<<<(N_E2E + B - 1) / B, B, 0, stream>>>(r, t, eei, mbuf, cnt);
    e2e_accum_k<<<(int)(((long)N_E2E * 32 + B - 1) / B), B, 0, stream>>>(g, r, t, eei, mbuf, denom, out);
    e2e_final_k<<<(N_EDGES * 32 + B - 1) / B, B, 0, stream>>>(denom, cnt, out);
}